// InteractiveBlock_39058432589929
// MI455X (gfx1250) — compile-verified
//
#include <hip/hip_runtime.h>
#include <hip/hip_bf16.h>
#include <math.h>

typedef __attribute__((ext_vector_type(2))) float v2f;
typedef __attribute__((ext_vector_type(8))) float v8f;

#define CE 512       // embedding channels
#define CC 256       // conv output channels
#define HP 196       // 14*14 valid positions
#define SM_STRIDE 208 // padded per-sample stride for `small` (16B-aligned, allows unguarded tile stores)
#define HWp 3136     // 56*56

// ---- Kernel 1: w_bar[c] = mean_o conv_w[o,c]; b_bar = mean(conv_b) ----
__global__ void prep_weights(const float* __restrict__ conv_w,
                             const float* __restrict__ conv_b,
                             float* __restrict__ wbar,
                             float* __restrict__ bbar) {
    int c = blockIdx.x * blockDim.x + threadIdx.x;
    if (c < CE) {
        float s = 0.0f;
        #pragma unroll 4
        for (int o = 0; o < CC; ++o) s += conv_w[(size_t)o * CE + c];
        wbar[c] = s * (1.0f / (float)CC);
    }
    if (blockIdx.x == 0 && threadIdx.x == 0) {
        float s = 0.0f;
        for (int o = 0; o < CC; ++o) s += conv_b[o];
        *bbar = s * (1.0f / (float)CC);
    }
}

// ---- Kernel 2: small[n,p] = sum_c emb[n,c,p] * wbar[c]  via V_WMMA_F32_16X16X4_F32 ----
// One wave (32 threads) per (sample, 16-position tile). D = A x B with
// A[M][K] = emb[n, kb+K, posbase+M]  (16x4 fp32, ISA layout: lanes0-15 K=0/1, lanes16-31 K=2/3)
// B[K][N] = wbar[kb+K] for all N     (every D column equals the matvec result)
__global__ void matvec_wmma(const float* __restrict__ emb,
                            const float* __restrict__ wbar,
                            float* __restrict__ small) {
    const int n       = blockIdx.y;
    const int posbase = blockIdx.x * 16;
    const int lane    = threadIdx.x;           // 0..31, EXEC all ones
    const int m       = lane & 15;             // M row handled by this lane's A regs
    const int khalf   = (lane >> 4) << 1;      // 0 for lanes 0-15, 2 for lanes 16-31

    unsigned p = (unsigned)(posbase + m);
    p = (p < HP) ? p : (HP - 1);               // arithmetic clamp (v_cndmask), no EXEC change

    const float* __restrict__ embn = emb + (size_t)n * CE * HP;

    v8f acc = {};
    for (int kb = 0; kb < CE; kb += 4) {
        v2f a, b;
        a.x = embn[(size_t)(kb + khalf)     * HP + p];
        a.y = embn[(size_t)(kb + khalf + 1) * HP + p];
        b.x = wbar[kb + khalf];
        b.y = wbar[kb + khalf + 1];
        acc = __builtin_amdgcn_wmma_f32_16x16x4_f32(
            /*neg_a=*/false, a, /*neg_b=*/false, b,
            /*c_mod=*/(short)0, acc, /*reuse_a=*/false, /*reuse_b=*/false);
    }

    // Column N=0 of D: lane 0 holds M=0..7 in acc[0..7], lane 16 holds M=8..15.
    // Stride is padded to 208, so the last tile's positions 196..207 land in this
    // sample's own padding -> store unconditionally, vectorized as two b128.
    if ((lane & 15) == 0) {
        float* __restrict__ outn = small + (size_t)n * SM_STRIDE + posbase + (lane >> 4) * 8;
        float4 lo, hi;
        lo.x = acc[0]; lo.y = acc[1]; lo.z = acc[2]; lo.w = acc[3];
        hi.x = acc[4]; hi.y = acc[5]; hi.z = acc[6]; hi.w = acc[7];
        *(float4*)(outn)     = lo;
        *(float4*)(outn + 4) = hi;
    }
}

// ---- Kernel 3: bilinear 14x14 -> 56x56 (half-pixel, edge-clamped), per-sample min-max norm ----
__global__ void upsample_norm(const float* __restrict__ small,
                              const float* __restrict__ bbar,
                              float* __restrict__ mask) {
    __shared__ float tile[HP];
    __shared__ float smn[256];
    __shared__ float smx[256];
    const int n   = blockIdx.x;
    const int tid = threadIdx.x;
    const float* __restrict__ sn = small + (size_t)n * SM_STRIDE;
    float* __restrict__ mk = mask + (size_t)n * HWp;

    for (int i = tid; i < HP; i += blockDim.x) tile[i] = sn[i];
    __syncthreads();

    const float bb = *bbar;
    float lmn =  INFINITY;
    float lmx = -INFINITY;
    for (int i = tid; i < HWp; i += blockDim.x) {
        int y = i / 56, x = i - y * 56;
        float sy = (y + 0.5f) * 0.25f - 0.5f;
        float sx = (x + 0.5f) * 0.25f - 0.5f;
        float fy = floorf(sy), fx = floorf(sx);
        float wy = sy - fy,    wx = sx - fx;
        int iy = (int)fy, ix = (int)fx;
        int y0 = max(0, min(13, iy)),     x0 = max(0, min(13, ix));
        int y1 = max(0, min(13, iy + 1)), x1 = max(0, min(13, ix + 1));
        float v00 = tile[y0 * 14 + x0], v01 = tile[y0 * 14 + x1];
        float v10 = tile[y1 * 14 + x0], v11 = tile[y1 * 14 + x1];
        float v = (1.0f - wy) * ((1.0f - wx) * v00 + wx * v01)
                +         wy  * ((1.0f - wx) * v10 + wx * v11) + bb;
        mk[i] = v;
        lmn = fminf(lmn, v);
        lmx = fmaxf(lmx, v);
    }
    smn[tid] = lmn; smx[tid] = lmx;
    __syncthreads();
    for (int s = blockDim.x >> 1; s > 0; s >>= 1) {
        if (tid < s) {
            smn[tid] = fminf(smn[tid], smn[tid + s]);
            smx[tid] = fmaxf(smx[tid], smx[tid + s]);
        }
        __syncthreads();
    }
    const float mn  = smn[0];
    const float inv = 1.0f / (smx[0] - smn[0]);
    for (int i = tid; i < HWp; i += blockDim.x)
        mk[i] = (mk[i] - mn) * inv;   // second min-max norm is the identity
}

// ---- Kernel 4: out = sigmoid(main) * (1 + mask*weight), float4 streaming (BW-bound) ----
__global__ void fuse(const float* __restrict__ main_in,
                     const float* __restrict__ mask,
                     const float* __restrict__ weight,
                     float* __restrict__ out,
                     long long n4) {
    long long i = (long long)blockIdx.x * blockDim.x + threadIdx.x;
    if (i >= n4) return;
    const float w = *weight;
    float4 v = ((const float4*)main_in)[i];
    long long flat = i * 4;
    long long n  = flat / ((long long)CC * HWp);
    long long hw = flat % HWp;                 // W=56 divisible by 4 -> aligned, same row
    float4 mkv = *(const float4*)(mask + n * HWp + hw);
    float4 r;
    float s;
    s = 1.0f / (1.0f + __expf(-v.x)); r.x = s * fmaf(mkv.x, w, 1.0f);
    s = 1.0f / (1.0f + __expf(-v.y)); r.y = s * fmaf(mkv.y, w, 1.0f);
    s = 1.0f / (1.0f + __expf(-v.z)); r.z = s * fmaf(mkv.z, w, 1.0f);
    s = 1.0f / (1.0f + __expf(-v.w)); r.w = s * fmaf(mkv.w, w, 1.0f);
    ((float4*)out)[i] = r;
}

extern "C" void kernel_launch(void* const* d_in, const int* in_sizes, int n_in,
                              void* d_out, int out_size, void* d_ws, size_t ws_size,
                              hipStream_t stream) {
    const float* main_in = (const float*)d_in[0];   // [32,256,56,56]
    const float* emb     = (const float*)d_in[1];   // [32,512,14,14]
    const float* conv_w  = (const float*)d_in[2];   // [256,512]
    const float* conv_b  = (const float*)d_in[3];   // [256]
    const float* weight  = (const float*)d_in[4];   // scalar
    float* out = (float*)d_out;

    float* ws    = (float*)d_ws;        // ~434 KB used
    float* wbar  = ws;                  // 512
    float* bbar  = ws + 512;            // 1
    float* small = ws + 1024;           // 32*208 (padded stride)
    float* mask  = ws + 8192;           // 32*3136

    prep_weights<<<2, 256, 0, stream>>>(conv_w, conv_b, wbar, bbar);
    matvec_wmma<<<dim3(13, 32), 32, 0, stream>>>(emb, wbar, small);   // 13 tiles x 32 samples
    upsample_norm<<<32, 256, 0, stream>>>(small, bbar, mask);
    long long n4 = (long long)out_size / 4;                            // 6,422,528
    fuse<<<(unsigned)((n4 + 255) / 256), 256, 0, stream>>>(main_in, mask, weight, out, n4);
}